// SimpleMaskRCNN_39041252720726
// MI455X (gfx1250) — compile-verified
//
#include <hip/hip_runtime.h>
#include <math.h>

#define NPROP 2000
#define NSORT 2048
#define KTOP  300
#define CCH   256
#define FH    128
#define FW    128
#define PP    7
#define NTAP  14
#define NT2   196      // 14*14 taps = K dimension
#define WROW  68       // LDS row stride for W (>=64, offset to avoid bank conflicts)
#define IMG_F 1024.0f

typedef float v2f __attribute__((ext_vector_type(2)));
typedef float v8f __attribute__((ext_vector_type(8)));

// ---------------- K1: decode + clip boxes ----------------
__global__ void k_decode(const float* __restrict__ prop, const float* __restrict__ delt,
                         float* __restrict__ boxes) {
    int n = blockIdx.x * blockDim.x + threadIdx.x;
    if (n >= NPROP) return;
    const float* p = prop + 4 * n;
    const float* d = delt + 4 * n;
    float pw  = fmaxf(p[2] - p[0], 1.0f);
    float ph  = fmaxf(p[3] - p[1], 1.0f);
    float pcx = p[0] + 0.5f * pw;
    float pcy = p[1] + 0.5f * ph;
    float dw  = fminf(d[2], 4.0f);
    float dh  = fminf(d[3], 4.0f);
    float ncx = pcx + d[0] * pw;
    float ncy = pcy + d[1] * ph;
    float nw  = pw * expf(dw);
    float nh  = ph * expf(dh);
    float b0 = ncx - 0.5f * nw, b1 = ncy - 0.5f * nh;
    float b2 = ncx + 0.5f * nw, b3 = ncy + 0.5f * nh;
    boxes[4*n+0] = fminf(fmaxf(b0, 0.0f), IMG_F);
    boxes[4*n+1] = fminf(fmaxf(b1, 0.0f), IMG_F);
    boxes[4*n+2] = fminf(fmaxf(b2, 0.0f), IMG_F);
    boxes[4*n+3] = fminf(fmaxf(b3, 0.0f), IMG_F);
}

// ---------------- K2: bitonic sort (descending score, stable by index) ----------------
__global__ void __launch_bounds__(1024) k_sort(const float* __restrict__ scores,
                                               const float* __restrict__ boxes,
                                               int* __restrict__ order,
                                               float* __restrict__ sboxes) {
    __shared__ float kk[NSORT];
    __shared__ int   vv[NSORT];
    int t = threadIdx.x;
    for (int e = t; e < NSORT; e += 1024) {
        kk[e] = (e < NPROP) ? scores[e] : -1e30f;
        vv[e] = e;
    }
    for (int size = 2; size <= NSORT; size <<= 1) {
        for (int stride = size >> 1; stride > 0; stride >>= 1) {
            __syncthreads();
            int pos = 2 * t - (t & (stride - 1));
            int j   = pos + stride;
            float k0 = kk[pos], k1 = kk[j];
            int   v0 = vv[pos], v1 = vv[j];
            // "elem0 precedes elem1" in final order: higher score first, tie -> lower index
            bool before = (k0 > k1) || (k0 == k1 && v0 < v1);
            bool dirAsc = ((pos & size) == 0);
            if (before != dirAsc) {
                kk[pos] = k1; kk[j] = k0;
                vv[pos] = v1; vv[j] = v0;
            }
        }
    }
    __syncthreads();
    for (int p = t; p < NPROP; p += 1024) {
        int id = vv[p];
        order[p] = id;
        sboxes[4*p+0] = boxes[4*id+0];
        sboxes[4*p+1] = boxes[4*id+1];
        sboxes[4*p+2] = boxes[4*id+2];
        sboxes[4*p+3] = boxes[4*id+3];
    }
}

// ---------------- K3: greedy NMS on sorted boxes ----------------
__global__ void __launch_bounds__(1024) k_nms(const float* __restrict__ sboxes,
                                              int* __restrict__ keep) {
    __shared__ float bx1[NPROP], by1[NPROP], bx2[NPROP], by2[NPROP];
    __shared__ int   su[NPROP];
    int t = threadIdx.x;
    for (int e = t; e < NPROP; e += 1024) {
        bx1[e] = sboxes[4*e+0];
        by1[e] = sboxes[4*e+1];
        bx2[e] = sboxes[4*e+2];
        by2[e] = sboxes[4*e+3];
        su[e] = 0;
    }
    for (int i = 0; i < NPROP - 1; ++i) {
        __syncthreads();
        if (su[i]) continue;                 // uniform across block
        float X1 = bx1[i], Y1 = by1[i], X2 = bx2[i], Y2 = by2[i];
        float ai = fmaxf(X2 - X1, 0.0f) * fmaxf(Y2 - Y1, 0.0f);
        for (int j = i + 1 + t; j < NPROP; j += 1024) {
            float aj  = fmaxf(bx2[j]-bx1[j], 0.0f) * fmaxf(by2[j]-by1[j], 0.0f);
            float ix1 = fmaxf(X1, bx1[j]);
            float iy1 = fmaxf(Y1, by1[j]);
            float ix2 = fminf(X2, bx2[j]);
            float iy2 = fminf(Y2, by2[j]);
            float inter = fmaxf(ix2 - ix1, 0.0f) * fmaxf(iy2 - iy1, 0.0f);
            float uni   = ai + aj - inter;
            if (inter / fmaxf(uni, 1e-6f) > 0.7f) su[j] = 1;
        }
    }
    __syncthreads();
    for (int e = t; e < NPROP; e += 1024) keep[e] = su[e] ? 0 : 1;
}

// ---------------- K4: compact first K kept boxes ----------------
__global__ void __launch_bounds__(512) k_compact(const int* __restrict__ keep,
                                                 const float* __restrict__ sboxes,
                                                 float* __restrict__ selb,
                                                 int* __restrict__ selv) {
    int t = threadIdx.x;
    if (t < KTOP) selv[t] = 0;
    __syncthreads();
    if (t == 0) {
        int cnt = 0;
        for (int p = 0; p < NPROP && cnt < KTOP; ++p) {
            if (keep[p]) {
                selb[4*cnt+0] = sboxes[4*p+0];
                selb[4*cnt+1] = sboxes[4*p+1];
                selb[4*cnt+2] = sboxes[4*p+2];
                selb[4*cnt+3] = sboxes[4*p+3];
                selv[cnt] = 1;
                ++cnt;
            }
        }
    }
}

// ---------------- K5: ROI-align via WMMA f32 16x16x4 ----------------
// Out[c][q] = sum_t G[c][t] * W[t][q], t = a*14 + b over (y-tap, x-tap) grid,
// W holds exactly the 4 bilinear weights per output q = j*7 + i.
// toff[] caches the per-tap feature byte offset so the k-loop has no div/mod.
__global__ void __launch_bounds__(512) k_roi(const float* __restrict__ feat,
                                             const float* __restrict__ selb,
                                             const int* __restrict__ selv,
                                             float* __restrict__ out) {
    __shared__ float Wlds[NT2 * WROW];
    __shared__ int   toff[NT2];          // (Yi[a]*FW + Xi[b]) * 4 bytes
    __shared__ int   Yi[NTAP], Xi[NTAP];
    __shared__ float Wx[PP], Wy[PP];

    int b = blockIdx.x;
    int t = threadIdx.x;
    float* obase = out + (size_t)b * (CCH * PP * PP);

    if (!selv[b]) {                           // uniform per block
        for (int e = t; e < CCH * PP * PP; e += 512) obase[e] = 0.0f;
        return;
    }

    // per-box sampling geometry (threads 0..6 -> x taps, 32..38 -> y taps)
    if (t < PP) {
        float x1 = selb[4*b+0] * 0.125f, x2 = selb[4*b+2] * 0.125f;
        float bw = fmaxf(x2 - x1, 0.001f);
        float xs = x1 + ((float)t + 0.5f) * bw * (1.0f / PP) - 0.5f;
        xs = fminf(fmaxf(xs, 0.0f), (float)(FW - 1));
        int x0 = (int)floorf(xs);
        Xi[t]      = x0;
        Xi[t + PP] = min(x0 + 1, FW - 1);
        Wx[t]      = xs - (float)x0;
    } else if (t >= 32 && t < 32 + PP) {
        int ty = t - 32;
        float y1 = selb[4*b+1] * 0.125f, y2 = selb[4*b+3] * 0.125f;
        float bh = fmaxf(y2 - y1, 0.001f);
        float ys = y1 + ((float)ty + 0.5f) * bh * (1.0f / PP) - 0.5f;
        ys = fminf(fmaxf(ys, 0.0f), (float)(FH - 1));
        int y0 = (int)floorf(ys);
        Yi[ty]      = y0;
        Yi[ty + PP] = min(y0 + 1, FH - 1);
        Wy[ty]      = ys - (float)y0;
    }
    // zero the (padded) weight matrix
    for (int e = t; e < NT2 * WROW; e += 512) Wlds[e] = 0.0f;
    __syncthreads();

    if (t < NT2) {
        // scatter the 49*4 bilinear weights: W[(j+7s)*14 + (i+7tt)][j*7+i]
        int j = t / 28, rem = t % 28;
        int i = rem / 4, c = rem % 4;
        int s = c >> 1, tt = c & 1;
        int row = (j + 7 * s) * NTAP + (i + 7 * tt);
        int col = j * PP + i;
        float wv = (s ? Wy[j] : 1.0f - Wy[j]) * (tt ? Wx[i] : 1.0f - Wx[i]);
        Wlds[row * WROW + col] = wv;
        // tap -> feature byte offset table (removes div/mod from the hot loop)
        int a = t / NTAP, bc = t % NTAP;
        toff[t] = (Yi[a] * FW + Xi[bc]) * 4;
    }
    __syncthreads();

    int  wave   = t >> 5;          // wave32
    int  lane   = t & 31;
    int  m      = lane & 15;
    bool hi     = lane >= 16;
    int  chbase = wave * 16;       // 16 waves x 16 channels = 256

    const char* fbyte  = (const char*)feat;
    int         chbyte = (chbase + m) * (FH * FW * 4);   // per-lane channel byte offset
    __builtin_prefetch(fbyte + chbyte + toff[0], 0, 0);  // global_prefetch_b8

    // per-lane-half bases: A/B VGPR0 uses tap e0 = 4k + (hi?2:0), VGPR1 uses e0+1
    const int*   toffp = toff + (hi ? 2 : 0);
    const float* wbase = &Wlds[(hi ? 2 : 0) * WROW + m];

    v8f acc0 = {}, acc1 = {}, acc2 = {}, acc3 = {};

    #pragma unroll
    for (int k = 0; k < 49; ++k) {
        int o0 = toffp[4 * k];
        int o1 = toffp[4 * k + 1];

        v2f av;                                  // A: M = channel (lane&15), K = e0/e1
        av.x = *(const float*)(fbyte + chbyte + o0);
        av.y = *(const float*)(fbyte + chbyte + o1);

        const float* wr = wbase + 4 * k * WROW;  // B: K = e0/e1, N = qbase + (lane&15)
        v2f b0v = { wr[0],  wr[WROW + 0]  };
        v2f b1v = { wr[16], wr[WROW + 16] };
        v2f b2v = { wr[32], wr[WROW + 32] };
        v2f b3v = { wr[48], wr[WROW + 48] };

        acc0 = __builtin_amdgcn_wmma_f32_16x16x4_f32(false, av, false, b0v, (short)0, acc0, false, false);
        acc1 = __builtin_amdgcn_wmma_f32_16x16x4_f32(false, av, false, b1v, (short)0, acc1, false, false);
        acc2 = __builtin_amdgcn_wmma_f32_16x16x4_f32(false, av, false, b2v, (short)0, acc2, false, false);
        acc3 = __builtin_amdgcn_wmma_f32_16x16x4_f32(false, av, false, b3v, (short)0, acc3, false, false);
    }

    // D layout: VGPR r -> M = r (lanes 0-15) / r+8 (lanes 16-31), N = lane&15
    #pragma unroll
    for (int nt = 0; nt < 4; ++nt) {
        v8f acc = (nt == 0) ? acc0 : (nt == 1) ? acc1 : (nt == 2) ? acc2 : acc3;
        int q = nt * 16 + m;
        if (q < PP * PP) {
            #pragma unroll
            for (int r = 0; r < 8; ++r) {
                int ch = chbase + r + (hi ? 8 : 0);
                obase[(size_t)ch * (PP * PP) + q] = acc[r];
            }
        }
    }
}

// ---------------- launch ----------------
extern "C" void kernel_launch(void* const* d_in, const int* in_sizes, int n_in,
                              void* d_out, int out_size, void* d_ws, size_t ws_size,
                              hipStream_t stream) {
    const float* feat = (const float*)d_in[0];   // (1,256,128,128)
    const float* prop = (const float*)d_in[1];   // (2000,4)
    const float* delt = (const float*)d_in[2];   // (2000,4)
    const float* scor = (const float*)d_in[3];   // (2000,1)
    float* out = (float*)d_out;                  // (300,256,7,7)

    float* boxes  = (float*)d_ws;                // 2000*4 f32
    float* sboxes = boxes + NPROP * 4;           // 2000*4 f32
    int*   order  = (int*)(sboxes + NPROP * 4);  // 2048 i32
    int*   keep   = order + NSORT;               // 2000 i32
    float* selb   = (float*)(keep + NPROP);      // 300*4 f32
    int*   selv   = (int*)(selb + KTOP * 4);     // 300 i32

    k_decode<<<(NPROP + 255) / 256, 256, 0, stream>>>(prop, delt, boxes);
    k_sort<<<1, 1024, 0, stream>>>(scor, boxes, order, sboxes);
    k_nms<<<1, 1024, 0, stream>>>(sboxes, keep);
    k_compact<<<1, 512, 0, stream>>>(keep, sboxes, selb, selv);
    k_roi<<<KTOP, 512, 0, stream>>>(feat, selb, selv, out);
}